// FullyConnectedTensorProduct_14791867367980
// MI455X (gfx1250) — compile-verified
//
#include <hip/hip_runtime.h>

// CDNA5 / gfx1250 fused O(3) tensor-product kernel (fp32 WMMA).
// out = alpha * [ Z(x1,x2) @ W ] with V_WMMA_F32_16X16X4_F32 tiles.
//
// Irreps 32x0e+32x1o on x1, x2, out. B = 8192.
//   x1/x2 row: [ s[0..31] | v[u][i] (u-major, i=0..2) ]   (128 f32)
//   weight: W[p][u][v][w], p in {0:ss->s, 1:sv->v, 2:vs->v, 3:vv->s}
//   out row: [ out0[w] | out1[w][j] (w-major) ]
//
// Per 16-row batch tile (one block, 4 waves):
//   wave 0   : out0   A0 = s1[u]*s2[v] (W0),  A3 = (v1[u].v2[v])/sqrt3 (W3)
//   wave 1..3: out1_j A1 = s1[u]*v2[v,j] (W1), A2 = v1[u,j]*s2[v] (W2)
// K is stepped 4-wide as (u, v..v+3); each wave issues 1024 WMMAs.
//
// LDS weight panel is stored in FRAGMENT-NATIVE pair layout:
//   chunk(p,ul)[ (v>>1)*64 + n*2 + (v&1) ] = W[p][u][v][n]
// so a B-fragment (two K-rows, same column) is ONE aligned ds_load_b64.

typedef float v2f __attribute__((ext_vector_type(2)));
typedef float v8f __attribute__((ext_vector_type(8)));

#define TP_MUL     32
#define TP_ROW     128
#define TP_PANEL_U 2
#define TP_PANEL_FLOATS (4 * TP_PANEL_U * TP_MUL * TP_MUL)   // 8192 f32 = 32 KB

__device__ __forceinline__ v8f wmma_f32_16x16x4(v2f a, v2f b, v8f c) {
    // (neg_a, A, neg_b, B, c_mod, C, reuse_a, reuse_b)
    return __builtin_amdgcn_wmma_f32_16x16x4_f32(false, a, false, b,
                                                 (short)0, c, false, false);
}

__global__ __launch_bounds__(128) void
tp_fused_wmma_kernel(const float* __restrict__ x1,
                     const float* __restrict__ x2,
                     const float* __restrict__ w,
                     float* __restrict__ out)
{
    extern __shared__ float smem[];              // 32 KB weight panel

    const int tid   = threadIdx.x;
    const int wave  = tid >> 5;                  // 0: out0 ; 1..3: out1 j=wave-1
    const int lane  = tid & 31;
    const int row   = lane & 15;                 // A-matrix M index
    const int khalf = lane >> 4;                 // 0: K=k,k+1 ; 1: K=k+2,k+3
    const int koff  = khalf * 2;
    const int ncol  = lane & 15;                 // column within an N-tile
    const int fragBase = khalf * 64 + ncol * 2;  // lane's f32 offset into a chunk

    const int tile  = blockIdx.x;
    const int grow  = tile * 16 + row;
    const float* __restrict__ x1r = x1 + (size_t)grow * TP_ROW;
    const float* __restrict__ x2r = x2 + (size_t)grow * TP_ROW;

    const float alpha = 2.2097086912079612e-02f;   // 1/sqrt(2*32*32)
    const float inv3  = 5.7735026918962584e-01f;   // 1/sqrt(3)
    const int   j     = wave - 1;

    // ---- per-lane register cache of u-invariant x2 data -------------------
    float s2a[8], s2b[8];                          // s2[v0], s2[v0+1], v0=4k+koff
    float v2xa[8], v2ya[8], v2za[8];               // v2[v0]   (wave 0)
    float v2xb[8], v2yb[8], v2zb[8];               // v2[v0+1] (wave 0)
    float vja[8], vjb[8];                          // v2[v0][j], v2[v0+1][j] (waves 1-3)

    if (wave == 0) {
        #pragma unroll
        for (int k = 0; k < 8; ++k) {
            const int v0 = 4 * k + koff;                 // even
            const v2f s2 = *(const v2f*)(x2r + v0);
            s2a[k] = s2.x; s2b[k] = s2.y;
            const float* p2 = x2r + 32 + 3 * v0;         // 8B aligned
            const v2f t0 = *(const v2f*)(p2);
            const v2f t1 = *(const v2f*)(p2 + 2);
            const v2f t2 = *(const v2f*)(p2 + 4);
            v2xa[k] = t0.x; v2ya[k] = t0.y; v2za[k] = t1.x;
            v2xb[k] = t1.y; v2yb[k] = t2.x; v2zb[k] = t2.y;
        }
    } else {
        #pragma unroll
        for (int k = 0; k < 8; ++k) {
            const int v0 = 4 * k + koff;
            const v2f s2 = *(const v2f*)(x2r + v0);
            s2a[k] = s2.x; s2b[k] = s2.y;
            vja[k] = x2r[32 + 3 * v0 + j];
            vjb[k] = x2r[32 + 3 * (v0 + 1) + j];
        }
    }

    v8f c0 = {};                                   // N-tile 0 (w = 0..15)
    v8f c1 = {};                                   // N-tile 1 (w = 16..31)

    for (int ub = 0; ub < TP_MUL; ub += TP_PANEL_U) {
        __syncthreads();
        // ---- cooperative transposing panel load: global -> LDS pair layout
        {
            const float4* __restrict__ gw4 = (const float4*)w;
            float4* l4 = (float4*)smem;
            #pragma unroll
            for (int i = 0; i < 8; ++i) {
                const int blk   = tid + i * 128;    // 2x4 transpose block id
                const int p_ul  = blk >> 7;         // 0..7 = (p, ul)
                const int r     = blk & 127;
                const int vpair = r >> 3;           // v = 2*vpair
                const int c4    = r & 7;            // cols 4*c4 .. 4*c4+3
                const int p     = p_ul >> 1;
                const int ul    = p_ul & 1;
                const size_t g  = ((size_t)(p * 32 + ub + ul) * 32 + 2 * vpair) * 8 + c4;
                const float4 r0 = gw4[g];           // row v
                const float4 r1 = gw4[g + 8];       // row v+1
                const int o = p_ul * 256 + vpair * 16 + c4 * 2;
                l4[o]     = make_float4(r0.x, r1.x, r0.y, r1.y);
                l4[o + 1] = make_float4(r0.z, r1.z, r0.w, r1.w);
            }
        }
        __syncthreads();

        // prefetch next panel into L2 while computing on this one
        if (ub + TP_PANEL_U < TP_MUL) {
            const char* nxt = (const char*)w
                            + ((size_t)wave * 32768 + (size_t)(ub + TP_PANEL_U) * 1024) * 4
                            + (size_t)lane * 256;
            __builtin_prefetch(nxt, 0, 1);
        }

        #pragma unroll
        for (int ul = 0; ul < TP_PANEL_U; ++ul) {
            const int u = ub + ul;
            if (wave == 0) {
                // ---------------- out0: paths 0 (W0) and 3 (W3) -------------
                const float s1u = x1r[u];
                const float v1x = x1r[32 + 3*u + 0];
                const float v1y = x1r[32 + 3*u + 1];
                const float v1z = x1r[32 + 3*u + 2];
                const float* __restrict__ wc0 = smem + (0*TP_PANEL_U + ul)*1024 + fragBase;
                const float* __restrict__ wc3 = smem + (3*TP_PANEL_U + ul)*1024 + fragBase;
                #pragma unroll
                for (int k = 0; k < 8; ++k) {        // vb = 4k
                    const v2f a0 = { s1u * s2a[k], s1u * s2b[k] };
                    const float d0 = v1x*v2xa[k] + v1y*v2ya[k] + v1z*v2za[k];
                    const float d1 = v1x*v2xb[k] + v1y*v2yb[k] + v1z*v2zb[k];
                    const v2f a3 = { inv3 * d0, inv3 * d1 };
                    const v2f b00 = *(const v2f*)(wc0 + k*128);
                    const v2f b01 = *(const v2f*)(wc0 + k*128 + 32);
                    const v2f b30 = *(const v2f*)(wc3 + k*128);
                    const v2f b31 = *(const v2f*)(wc3 + k*128 + 32);
                    c0 = wmma_f32_16x16x4(a0, b00, c0);
                    c1 = wmma_f32_16x16x4(a0, b01, c1);
                    c0 = wmma_f32_16x16x4(a3, b30, c0);
                    c1 = wmma_f32_16x16x4(a3, b31, c1);
                }
            } else {
                // ------------- out1 (component j): paths 1 (W1), 2 (W2) -----
                const float s1u  = x1r[u];
                const float v1uj = x1r[32 + 3*u + j];
                const float* __restrict__ wc1 = smem + (1*TP_PANEL_U + ul)*1024 + fragBase;
                const float* __restrict__ wc2 = smem + (2*TP_PANEL_U + ul)*1024 + fragBase;
                #pragma unroll
                for (int k = 0; k < 8; ++k) {
                    const v2f a1 = { s1u  * vja[k], s1u  * vjb[k] };
                    const v2f a2 = { v1uj * s2a[k], v1uj * s2b[k] };
                    const v2f b10 = *(const v2f*)(wc1 + k*128);
                    const v2f b11 = *(const v2f*)(wc1 + k*128 + 32);
                    const v2f b20 = *(const v2f*)(wc2 + k*128);
                    const v2f b21 = *(const v2f*)(wc2 + k*128 + 32);
                    c0 = wmma_f32_16x16x4(a1, b10, c0);
                    c1 = wmma_f32_16x16x4(a1, b11, c1);
                    c0 = wmma_f32_16x16x4(a2, b20, c0);
                    c1 = wmma_f32_16x16x4(a2, b21, c1);
                }
            }
        }
    }

    // ---- epilogue: C layout VGPR r -> (M = r + 8*khalf, N = ncol [+16]) ----
    const int mbase = khalf * 8;
    if (wave == 0) {
        #pragma unroll
        for (int r = 0; r < 8; ++r) {
            const int m = mbase + r;
            float* o = out + (size_t)(tile*16 + m) * TP_ROW;
            o[ncol]      = alpha * c0[r];
            o[16 + ncol] = alpha * c1[r];
        }
    } else {
        #pragma unroll
        for (int r = 0; r < 8; ++r) {
            const int m = mbase + r;
            float* o = out + (size_t)(tile*16 + m) * TP_ROW + 32;
            o[ncol*3 + j]        = alpha * c0[r];   // out1[w=ncol,    j]
            o[(16 + ncol)*3 + j] = alpha * c1[r];   // out1[w=16+ncol, j]
        }
    }
}

extern "C" void kernel_launch(void* const* d_in, const int* in_sizes, int n_in,
                              void* d_out, int out_size, void* d_ws, size_t ws_size,
                              hipStream_t stream) {
    (void)n_in; (void)out_size; (void)d_ws; (void)ws_size;
    const float* x1 = (const float*)d_in[0];
    const float* x2 = (const float*)d_in[1];
    const float* w  = (const float*)d_in[2];
    float* out      = (float*)d_out;

    const int B = in_sizes[0] / TP_ROW;     // 8192
    dim3 grid(B / 16), block(128);          // 4 waves: out0, out1 j=0..2
    size_t lds_bytes = TP_PANEL_FLOATS * sizeof(float);   // 32 KB
    tp_fused_wmma_kernel<<<grid, block, lds_bytes, stream>>>(x1, x2, w, out);
}